// EnhancedGNNModel_19705309954560
// MI455X (gfx1250) — compile-verified
//
#include <hip/hip_runtime.h>

typedef float v2f __attribute__((ext_vector_type(2)));
typedef float v8f __attribute__((ext_vector_type(8)));

#define GCN_N 50000
#define GCN_K 128

// ---------------------------------------------------------------------------
// Degree / normalization
// ---------------------------------------------------------------------------
__global__ void deg_init_kernel(float* deg, int n) {
    int i = blockIdx.x * blockDim.x + threadIdx.x;
    if (i < n) deg[i] = 1.0f;  // self-loop contributes 1
}

__global__ void deg_edges_kernel(const int* __restrict__ dst, float* deg, int e) {
    int i = blockIdx.x * blockDim.x + threadIdx.x;
    if (i < e) atomicAdd(&deg[dst[i]], 1.0f);
}

__global__ void deg_to_dinv_kernel(float* deg, int n) {
    int i = blockIdx.x * blockDim.x + threadIdx.x;
    if (i < n) deg[i] = rsqrtf(deg[i]);  // deg >= 1 always (self-loop)
}

// ---------------------------------------------------------------------------
// f32 WMMA GEMM:  C[nrows x (NT*16)] = X[nrows x 128] @ W[128 x (NT*16)]
// One wave computes a 16-row strip across all NT 16-wide tiles, reusing the
// A fragment. Layout per CDNA5 ISA 7.12.2 (16x16x4 f32 shape):
//   A (16x4):  vgpr v, lane half h -> K = 2*h + v, M = lane&15
//   B (4x16):  vgpr v, lane half h -> K = 2*h + v, N = lane&15
//   C (16x16): vgpr v, lane half h -> M = v + 8*h, N = lane&15
// ---------------------------------------------------------------------------
template <int NT>
__global__ void gemm_wmma_f32_kernel(const float* __restrict__ X,
                                     const float* __restrict__ W,
                                     float* __restrict__ C,
                                     int mtiles) {
    const int lane = threadIdx.x & 31;
    const int wave = blockIdx.x * (blockDim.x >> 5) + (threadIdx.x >> 5);
    if (wave >= mtiles) return;               // wave-uniform: EXEC stays all-1s
    const int m0   = wave << 4;
    const int half = lane >> 4;               // 0 or 1
    const int r    = lane & 15;
    const int F    = NT * 16;

    v8f acc[NT] = {};

    const float* xrow = X + (size_t)(m0 + r) * GCN_K;
    for (int k = 0; k < GCN_K; k += 4) {
        const int ka = k + 2 * half;
        v2f a;
        a.x = xrow[ka];
        a.y = xrow[ka + 1];
#pragma unroll
        for (int t = 0; t < NT; ++t) {
            v2f b;
            b.x = W[(size_t)ka * F + t * 16 + r];
            b.y = W[(size_t)(ka + 1) * F + t * 16 + r];
            acc[t] = __builtin_amdgcn_wmma_f32_16x16x4_f32(
                false, a, false, b, (short)0, acc[t], false, false);
        }
    }

#pragma unroll
    for (int t = 0; t < NT; ++t) {
#pragma unroll
        for (int v = 0; v < 8; ++v) {
            C[(size_t)(m0 + v + 8 * half) * F + t * 16 + r] = acc[t][v];
        }
    }
}

// ---------------------------------------------------------------------------
// Aggregation init: out[n,f] = xw[n,f] * dinv[n]^2 + bias[f]   (self-loop term)
// Vectorized: one float4 per thread. F is a multiple of 4.
// ---------------------------------------------------------------------------
template <int F>
__global__ void agg_init_kernel(const float4* __restrict__ xw,
                                const float* __restrict__ dinv,
                                const float4* __restrict__ bias,
                                float4* __restrict__ out, int nvec) {
    int idx = blockIdx.x * blockDim.x + threadIdx.x;
    if (idx >= nvec) return;
    constexpr int FV = F / 4;
    int node = idx / FV;
    int f4   = idx - node * FV;
    float di = dinv[node];
    float s  = di * di;
    float4 v = xw[idx];
    float4 b = bias[f4];
    float4 o;
    o.x = v.x * s + b.x;
    o.y = v.y * s + b.y;
    o.z = v.z * s + b.z;
    o.w = v.w * s + b.w;
    out[idx] = o;
}

// ---------------------------------------------------------------------------
// Edge aggregation: out[dst] += xw[src] * (dinv[src]*dinv[dst]), one wave/edge.
// VEC = F/32 floats per lane: one vector gather (b128/b64) + VEC atomics.
// ---------------------------------------------------------------------------
template <int VEC>
__global__ void agg_edges_kernel(const float* __restrict__ xw,
                                 const int* __restrict__ src,
                                 const int* __restrict__ dst,
                                 const float* __restrict__ dinv,
                                 float* __restrict__ out, int e) {
    const int F = VEC * 32;
    int wave = blockIdx.x * (blockDim.x >> 5) + (threadIdx.x >> 5);
    int lane = threadIdx.x & 31;
    if (wave >= e) return;
    int s = src[wave];
    int d = dst[wave];
    float w = dinv[s] * dinv[d];
    const float* xs = xw + (size_t)s * F + lane * VEC;
    float*       od = out + (size_t)d * F + lane * VEC;

    float m[VEC];
    if (VEC == 4) {
        float4 v = *(const float4*)xs;
        m[0] = v.x; m[1] = v.y; m[2] = v.z; m[3] = v.w;
    } else {
        float2 v = *(const float2*)xs;
        m[0] = v.x; m[1] = v.y;
    }
#pragma unroll
    for (int j = 0; j < VEC; ++j)
        atomicAdd(&od[j], m[j] * w);
}

// ---------------------------------------------------------------------------
// Fused ReLU + LayerNorm over F=128 features, one wave per node (wave32)
// ---------------------------------------------------------------------------
__global__ void relu_ln_kernel(float* __restrict__ h,
                               const float* __restrict__ g,
                               const float* __restrict__ be, int n) {
    int wave = blockIdx.x * (blockDim.x >> 5) + (threadIdx.x >> 5);
    int lane = threadIdx.x & 31;
    if (wave >= n) return;
    float4* row = (float4*)(h + (size_t)wave * 128);
    float4 v = row[lane];
    v.x = fmaxf(v.x, 0.0f); v.y = fmaxf(v.y, 0.0f);
    v.z = fmaxf(v.z, 0.0f); v.w = fmaxf(v.w, 0.0f);

    float s = v.x + v.y + v.z + v.w;
#pragma unroll
    for (int m = 16; m >= 1; m >>= 1) s += __shfl_xor(s, m, 32);
    float mu = s * (1.0f / 128.0f);

    float dx = v.x - mu, dy = v.y - mu, dz = v.z - mu, dw = v.w - mu;
    float q = dx * dx + dy * dy + dz * dz + dw * dw;
#pragma unroll
    for (int m = 16; m >= 1; m >>= 1) q += __shfl_xor(q, m, 32);
    float rs = rsqrtf(q * (1.0f / 128.0f) + 1e-5f);

    const float4 gg = ((const float4*)g)[lane];
    const float4 bb = ((const float4*)be)[lane];
    float4 o;
    o.x = dx * rs * gg.x + bb.x;
    o.y = dy * rs * gg.y + bb.y;
    o.z = dz * rs * gg.z + bb.z;
    o.w = dw * rs * gg.w + bb.w;
    row[lane] = o;
}

// ---------------------------------------------------------------------------
// Launch
// ---------------------------------------------------------------------------
extern "C" void kernel_launch(void* const* d_in, const int* in_sizes, int n_in,
                              void* d_out, int out_size, void* d_ws, size_t ws_size,
                              hipStream_t stream) {
    const float* x   = (const float*)d_in[0];
    const int*   ei  = (const int*)d_in[1];
    const float* W1  = (const float*)d_in[2];
    const float* b1  = (const float*)d_in[3];
    const float* W2  = (const float*)d_in[4];
    const float* b2  = (const float*)d_in[5];
    const float* W3  = (const float*)d_in[6];
    const float* b3  = (const float*)d_in[7];
    const float* g1  = (const float*)d_in[8];
    const float* be1 = (const float*)d_in[9];
    const float* g2  = (const float*)d_in[10];
    const float* be2 = (const float*)d_in[11];

    const int N = GCN_N;
    const int E = in_sizes[1] / 2;
    const int* src = ei;
    const int* dst = ei + E;

    // Workspace layout
    char*  ws   = (char*)d_ws;
    float* dinv = (float*)ws;                                   // N floats
    size_t off  = ((size_t)N * sizeof(float) + 255) & ~(size_t)255;
    float* bufA = (float*)(ws + off);                           // N*128 floats
    float* bufB = bufA + (size_t)N * 128;                       // N*128 floats

    const int T256 = 256;
    const int mtiles = N / 16;                                  // 3125
    const int gemmBlocks = (mtiles + 7) / 8;                    // 8 waves/block
    const int edgeWaveBlocks = (E + 7) / 8;                     // 8 edges/block
    const int nodeWaveBlocks = (N + 7) / 8;
    const int v128 = N * (128 / 4);                             // float4 count
    const int v64  = N * (64 / 4);

    // ---- normalization ----
    deg_init_kernel<<<(N + T256 - 1) / T256, T256, 0, stream>>>(dinv, N);
    deg_edges_kernel<<<(E + T256 - 1) / T256, T256, 0, stream>>>(dst, dinv, E);
    deg_to_dinv_kernel<<<(N + T256 - 1) / T256, T256, 0, stream>>>(dinv, N);

    // ---- layer 1: h1 = LN(relu(agg(x@W1) + b1)) ----
    gemm_wmma_f32_kernel<8><<<gemmBlocks, T256, 0, stream>>>(x, W1, bufA, mtiles);
    agg_init_kernel<128><<<(v128 + T256 - 1) / T256, T256, 0, stream>>>(
        (const float4*)bufA, dinv, (const float4*)b1, (float4*)bufB, v128);
    agg_edges_kernel<4><<<edgeWaveBlocks, T256, 0, stream>>>(bufA, src, dst, dinv, bufB, E);
    relu_ln_kernel<<<nodeWaveBlocks, T256, 0, stream>>>(bufB, g1, be1, N);

    // ---- layer 2: h2 = LN(relu(agg(h1@W2) + b2)) ----
    gemm_wmma_f32_kernel<8><<<gemmBlocks, T256, 0, stream>>>(bufB, W2, bufA, mtiles);
    agg_init_kernel<128><<<(v128 + T256 - 1) / T256, T256, 0, stream>>>(
        (const float4*)bufA, dinv, (const float4*)b2, (float4*)bufB, v128);
    agg_edges_kernel<4><<<edgeWaveBlocks, T256, 0, stream>>>(bufA, src, dst, dinv, bufB, E);
    relu_ln_kernel<<<nodeWaveBlocks, T256, 0, stream>>>(bufB, g2, be2, N);

    // ---- layer 3: out = agg(h2@W3) + b3 (written straight into d_out) ----
    float* out = (float*)d_out;
    gemm_wmma_f32_kernel<4><<<gemmBlocks, T256, 0, stream>>>(bufB, W3, bufA, mtiles);
    agg_init_kernel<64><<<(v64 + T256 - 1) / T256, T256, 0, stream>>>(
        (const float4*)bufA, dinv, (const float4*)b3, (float4*)out, v64);
    agg_edges_kernel<2><<<edgeWaveBlocks, T256, 0, stream>>>(bufA, src, dst, dinv, out, E);
}